// AttentionEssential_51238959841469
// MI455X (gfx1250) — compile-verified
//
#include <hip/hip_runtime.h>
#include <hip/hip_bf16.h>
#include <stdint.h>
#include <stddef.h>

#define S_LEN     2048
#define W_STRIDE  4096   // my_attention_mask row stride (2*S floats)
#define MASK_ID   103
#define NTHREADS  256
#define EPT       8      // elements per thread (256*8 = 2048)

// ---------------- deterministic counter-based RNG (murmur3-style) ----------
__device__ __forceinline__ unsigned mixu(unsigned h) {
  h ^= h >> 16; h *= 0x85EBCA6Bu;
  h ^= h >> 13; h *= 0xC2B2AE35u;
  h ^= h >> 16; return h;
}
__device__ __forceinline__ unsigned hash3(unsigned a, unsigned b, unsigned c) {
  unsigned h = a * 0x9E3779B1u + 0x165667B1u;
  h = mixu(h ^ (b * 0x85EBCA77u));
  h = mixu(h ^ (c * 0xC2B2AE3Du));
  return h;
}
__device__ __forceinline__ float u01(unsigned h) {
  // strictly inside (0,1) so log/loglog are finite
  return ((float)(h >> 8) + 0.5f) * (1.0f / 16777216.0f);
}

__global__ __launch_bounds__(NTHREADS)
void gumbel_topk_mask_kernel(const float* __restrict__ w_all,
                             const int*   __restrict__ am_all,
                             const int*   __restrict__ ids_all,
                             int*         __restrict__ out,
                             long long nrows) {
  __shared__ __align__(16) int ids_lds[S_LEN];   // async-staged input_ids row
  __shared__ int hist[256];
  __shared__ int sfx[256];
  __shared__ int shSelByte;
  __shared__ int shSelGt;
  __shared__ int shNumValid;
  __shared__ int shK;

  const unsigned r   = blockIdx.x;
  const unsigned tid = threadIdx.x;
  const unsigned e0  = tid * EPT;

  // ---- 1) async-stage input_ids row into LDS (overlaps with key compute) ----
  // GVS mode: mem = SADDR(64b sgpr) + VADDR(32b byte offset) + inst_offset,
  //           lds = VDST(lds byte addr) + inst_offset.  Tracked by ASYNCcnt.
  {
    const int* idRow = ids_all + (size_t)r * S_LEN;
    unsigned ldsBase = (unsigned)(size_t)(&ids_lds[0]) + tid * 16u; // low 32 bits of flat = LDS offset
    unsigned memOff  = tid * 16u;
    asm volatile("global_load_async_to_lds_b128 %0, %1, %2"
                 :: "v"(ldsBase), "v"(memOff), "s"(idRow) : "memory");
    asm volatile("global_load_async_to_lds_b128 %0, %1, %2 offset:4096"
                 :: "v"(ldsBase), "v"(memOff), "s"(idRow) : "memory");
  }

  // ---- 2) Gumbel keys -> sortable uint32, kept in registers ----
  unsigned sk[EPT];
  {
    const float*  wrow = w_all + (size_t)r * W_STRIDE;   // only first S of row used
    const float4* w4   = reinterpret_cast<const float4*>(wrow + e0);
    float4 wa = w4[0], wb = w4[1];
    float wv[EPT] = {wa.x, wa.y, wa.z, wa.w, wb.x, wb.y, wb.z, wb.w};
#pragma unroll
    for (int j = 0; j < EPT; ++j) {
      float x  = wv[j];
      float lw = (x > 0.0f) ? __logf(fmaxf(x, 1e-30f)) : -__builtin_inff();
      float u  = u01(hash3(0xC0FFEEu, r, e0 + (unsigned)j));
      float g  = -__logf(-__logf(u));          // Gumbel(0,1)
      float key = lw + g;                       // -inf stays -inf (zero weight)
      unsigned ub = __float_as_uint(key);
      sk[j] = (ub >> 31) ? ~ub : (ub | 0x80000000u); // monotone: bigger float -> bigger uint
    }
  }

  // ---- 3) row-sum of attention_mask, derive k = floor(sum * frac) ----
  {
    const int4* m4 = reinterpret_cast<const int4*>(am_all + (size_t)r * S_LEN + e0);
    int4 ma = m4[0], mb = m4[1];
    int lsum = ma.x + ma.y + ma.z + ma.w + mb.x + mb.y + mb.z + mb.w;
    if (tid == 0) shNumValid = 0;
    __syncthreads();
    atomicAdd(&shNumValid, lsum);
    __syncthreads();
    if (tid == 0) {
      float u1  = u01(hash3(42u, r, 11u));
      float u2  = u01(hash3(42u, r, 22u));
      float nrm = sqrtf(-2.0f * __logf(u1)) * __cosf(6.28318530718f * u2);
      float frac = 0.15f + 0.0375f * nrm;       // MU_P + sigma*N(0,1), sigma=min(.05,.0375)
      int k = (int)floorf((float)shNumValid * frac);
      shK = k < 0 ? 0 : (k > S_LEN ? S_LEN : k);
    }
    __syncthreads();
  }
  const int k = shK;         // uniform across block

  // ---- 4) exact top-k threshold via 4-pass MSB radix select ----
  unsigned Tval = 0;
  int remFinal  = 0;
  const bool doSel = (k > 0);  // uniform branch: barrier structure stays legal
  if (doSel) {
    unsigned prefix = 0;
    int remaining   = k;
    for (int p = 3; p >= 0; --p) {
      hist[tid] = 0;
      __syncthreads();
      const unsigned hm = (p == 3) ? 0u : (0xFFFFFFFFu << ((unsigned)(p + 1) * 8u));
#pragma unroll
      for (int j = 0; j < EPT; ++j) {
        unsigned s = sk[j];
        if ((s & hm) == prefix)
          atomicAdd(&hist[(s >> (p * 8)) & 0xFFu], 1);   // ds_add_u32
      }
      __syncthreads();
      sfx[tid] = hist[tid];
      __syncthreads();
      // inclusive suffix-sum over 256 bins (Hillis-Steele)
      for (int off = 1; off < 256; off <<= 1) {
        int add = (tid + (unsigned)off < 256u) ? sfx[tid + off] : 0;
        __syncthreads();
        sfx[tid] += add;
        __syncthreads();
      }
      int gt = (tid < 255u) ? sfx[tid + 1] : 0;          // count strictly above this bin
      if (remaining > gt && remaining <= sfx[tid]) { shSelByte = (int)tid; shSelGt = gt; }
      __syncthreads();
      prefix    |= ((unsigned)shSelByte) << (p * 8);
      remaining -= shSelGt;
      __syncthreads();
    }
    Tval     = prefix;
    remFinal = remaining;    // how many == T elements to take, by index order
  }

  // ---- 5) tie-break by index: cross-thread prefix of equal-to-T counts ----
  int selFlag[EPT];
  if (doSel) {
    int eqoff[EPT];
    int cnt = 0;
#pragma unroll
    for (int j = 0; j < EPT; ++j) { eqoff[j] = cnt; cnt += (sk[j] == Tval) ? 1 : 0; }
    sfx[tid] = cnt;
    __syncthreads();
    for (int off = 1; off < 256; off <<= 1) {
      int add = (tid >= (unsigned)off) ? sfx[tid - off] : 0;
      __syncthreads();
      sfx[tid] += add;
      __syncthreads();
    }
    int excl = sfx[tid] - cnt;
#pragma unroll
    for (int j = 0; j < EPT; ++j) {
      unsigned s = sk[j];
      selFlag[j] = (s > Tval) || (s == Tval && (excl + eqoff[j]) < remFinal);
    }
  } else {
#pragma unroll
    for (int j = 0; j < EPT; ++j) selFlag[j] = 0;
  }

  // ---- 6) wait for async ids staging, then vectorized int4 output stores ----
  asm volatile("s_wait_asynccnt 0" ::: "memory");  // my wave's async loads done
  __syncthreads();                                  // everyone's loads visible

  const long long rowBase = (long long)r * S_LEN;
  int* oid = out + rowBase;
  int* omk = out + (long long)nrows * S_LEN + rowBase;
  int* olb = out + 2LL * (long long)nrows * S_LEN + rowBase;

  int oi[EPT], om[EPT], ol[EPT];
#pragma unroll
  for (int j = 0; j < EPT; ++j) {
    int m = selFlag[j];
    om[j] = m;
    ol[j] = -m;
    oi[j] = m ? MASK_ID : ids_lds[e0 + j];
  }
  reinterpret_cast<int4*>(oid + e0)[0] = make_int4(oi[0], oi[1], oi[2], oi[3]);
  reinterpret_cast<int4*>(oid + e0)[1] = make_int4(oi[4], oi[5], oi[6], oi[7]);
  reinterpret_cast<int4*>(omk + e0)[0] = make_int4(om[0], om[1], om[2], om[3]);
  reinterpret_cast<int4*>(omk + e0)[1] = make_int4(om[4], om[5], om[6], om[7]);
  reinterpret_cast<int4*>(olb + e0)[0] = make_int4(ol[0], ol[1], ol[2], ol[3]);
  reinterpret_cast<int4*>(olb + e0)[1] = make_int4(ol[4], ol[5], ol[6], ol[7]);
}

extern "C" void kernel_launch(void* const* d_in, const int* in_sizes, int n_in,
                              void* d_out, int out_size, void* d_ws, size_t ws_size,
                              hipStream_t stream) {
  const float* w   = (const float*)d_in[0];  // my_attention_mask (B,C,2S) f32
  const int*   am  = (const int*)d_in[1];    // attention_mask    (B,C,S)  i32
  const int*   ids = (const int*)d_in[2];    // input_ids         (B,C,S)  i32
  int* out = (int*)d_out;                    // 3 concatenated (B,C,S) i32 outputs

  long long nrows = (long long)in_sizes[1] / S_LEN;   // B*C rows
  dim3 grid((unsigned)nrows), block(NTHREADS);
  hipLaunchKernelGGL(gumbel_topk_mask_kernel, grid, block, 0, stream,
                     w, am, ids, out, nrows);
}